// PlaceCellNetwork_11244224381333
// MI455X (gfx1250) — compile-verified
//
#include <hip/hip_runtime.h>

// Problem constants (from reference)
#define NROWS 262144
#define DIN   3
#define DOUT  20
#define NSTEPS 100
#define DT_C    0.1f
#define ALPHA_C 1.0f
#define LBD1_C  0.0f
#define LBD2_C  0.0f
#define TOL_C   1e-4f

// Phase-1 geometry: 256 blocks x 8 waves x 8 row-groups x 16 rows = 262144 rows
#define PH1_BLOCKS 256
#define PH1_WAVES  8
#define PH1_GROUPS 8
// Phase-3 geometry: 2048 blocks x 8 waves x 16 rows
#define PH3_BLOCKS 2048

#define PART_FLOATS (PH1_BLOCKS * NSTEPS * 2)

typedef float v2f __attribute__((ext_vector_type(2)));
typedef float v8f __attribute__((ext_vector_type(8)));

// Build per-lane constants: B fragments for V_WMMA_F32_16X16X4_F32 (W^T slices),
// bias values, diag(M) and reciprocal denominators. All loads unconditional
// (clamped index) so no exec-divergent load branches are generated.
// ISA layouts (wave32):
//   A (16x4, MxK):  lanes 0-15 hold {K0,K1}, lanes 16-31 hold {K2,K3}; M = lane&15
//   B (4x16, KxN):  lanes 0-15 hold {K0,K1}, lanes 16-31 hold {K2,K3}; N = lane&15
//   C/D (16x16):    reg r -> M = r + 8*(lane>>4), N = lane&15
__device__ __forceinline__ void make_lane_consts(const float* __restrict__ W,
                                                 const float* __restrict__ Mm,
                                                 const float* __restrict__ bv,
                                                 int lane,
                                                 v2f& b0, v2f& b1,
                                                 float& bb0, float& bb1,
                                                 float& dm0, float& dm1,
                                                 float& rden0, float& rden1) {
    const int half = lane >> 4;
    const int l15  = lane & 15;
    const int n0   = l15;
    const int n1   = l15 + 16;
    const bool v1  = (n1 < DOUT);
    const int n1c  = v1 ? n1 : 0;  // clamped index -> unconditional loads

    const float w00 = W[n0 * DIN + 0], w01 = W[n0 * DIN + 1], w02 = W[n0 * DIN + 2];
    const float t10 = W[n1c * DIN + 0], t11 = W[n1c * DIN + 1], t12 = W[n1c * DIN + 2];
    const float w10 = v1 ? t10 : 0.0f;
    const float w11 = v1 ? t11 : 0.0f;
    const float w12 = v1 ? t12 : 0.0f;

    b0.x = half ? w02 : w00;  b0.y = half ? 0.0f : w01;
    b1.x = half ? w12 : w10;  b1.y = half ? 0.0f : w11;

    bb0 = ALPHA_C * bv[n0];
    const float tb1 = bv[n1c];
    bb1 = v1 ? (ALPHA_C * tb1) : 0.0f;

    dm0 = Mm[n0 * DOUT + n0];
    const float tm1 = Mm[n1c * DOUT + n1c];
    dm1 = v1 ? tm1 : 0.0f;   // invalid lanes: drive=0, dm=0, rden=1 => state stays 0

    // Hoisted reciprocal: one exact division at setup, multiply in the hot loop.
    rden0 = 1.0f / (LBD2_C + dm0);
    rden1 = v1 ? (1.0f / (LBD2_C + tm1)) : 1.0f;
}

// drive tiles for 16 rows starting at rowBase (loads X, 2x WMMA, bias subtract)
__device__ __forceinline__ void compute_drive(const float* __restrict__ X,
                                              int rowBase, int lane,
                                              v2f b0, v2f b1, float bb0, float bb1,
                                              v8f& d0, v8f& d1) {
    const int half = lane >> 4;
    const int l15  = lane & 15;
    const int row  = rowBase + l15;

    const float x0 = X[row * DIN + 0];
    const float x1 = X[row * DIN + 1];
    const float x2 = X[row * DIN + 2];
    v2f a;
    a.x = half ? x2 : x0;
    a.y = half ? 0.0f : x1;

    v8f c0 = {};
    v8f c1 = {};
    // D = A x B + C ; args: (neg_a, A, neg_b, B, c_mod, C, reuse_a, reuse_b)
    c0 = __builtin_amdgcn_wmma_f32_16x16x4_f32(false, a, false, b0, (short)0, c0, false, false);
    c1 = __builtin_amdgcn_wmma_f32_16x16x4_f32(false, a, false, b1, (short)0, c1, false, false);

#pragma unroll
    for (int r = 0; r < 8; ++r) {
        c0[r] -= bb0;
        c1[r] -= bb1;
    }
    d0 = c0;
    d1 = c1;
}

// Phase 1: run the full unfrozen trajectory, accumulate per-step sum((Ynew-Yold)^2)
// and sum(Yold^2). Deterministic reductions only (shfl butterfly + fixed-order LDS).
__global__ __launch_bounds__(256) void pcn_phase1(const float* __restrict__ X,
                                                  const float* __restrict__ W,
                                                  const float* __restrict__ Mm,
                                                  const float* __restrict__ bv,
                                                  float* __restrict__ part) {
    __shared__ float acc[PH1_WAVES][NSTEPS][2];
    const int tid = threadIdx.x;
    for (int i = tid; i < PH1_WAVES * NSTEPS * 2; i += 256)
        ((float*)acc)[i] = 0.0f;
    __syncthreads();

    const int lane = tid & 31;
    const int wave = tid >> 5;

    v2f b0, b1;
    float bb0, bb1, dm0, dm1, rden0, rden1;
    make_lane_consts(W, Mm, bv, lane, b0, b1, bb0, bb1, dm0, dm1, rden0, rden1);

    for (int g = 0; g < PH1_GROUPS; ++g) {
        const int rowBase = ((blockIdx.x * PH1_WAVES + wave) * PH1_GROUPS + g) * 16;
        v8f d0, d1;
        compute_drive(X, rowBase, lane, b0, b1, bb0, bb1, d0, d1);

        float uy0[8], yo0[8], uy1[8], yo1[8];
#pragma unroll
        for (int r = 0; r < 8; ++r) { uy0[r] = 0.f; yo0[r] = 0.f; uy1[r] = 0.f; yo1[r] = 0.f; }

        for (int s = 0; s < NSTEPS; ++s) {
            float ds = 0.0f, ys = 0.0f;
#pragma unroll
            for (int r = 0; r < 8; ++r) {
                float du0 = d0[r] - uy0[r] - yo0[r] * dm0;
                uy0[r] += DT_C * du0;
                float yn0 = fmaxf((uy0[r] - LBD1_C) * rden0, 0.0f);
                float df0 = yn0 - yo0[r];
                ds += df0 * df0;
                ys += yo0[r] * yo0[r];
                yo0[r] = yn0;

                // invalid lanes have d1=0, dm1=0, rden1=1 -> state stays 0,
                // contributes exactly 0 to the sums: no guard needed.
                float du1 = d1[r] - uy1[r] - yo1[r] * dm1;
                uy1[r] += DT_C * du1;
                float yn1 = fmaxf((uy1[r] - LBD1_C) * rden1, 0.0f);
                float df1 = yn1 - yo1[r];
                ds += df1 * df1;
                ys += yo1[r] * yo1[r];
                yo1[r] = yn1;
            }
            // wave32 butterfly reduction (deterministic tree)
#pragma unroll
            for (int m = 16; m >= 1; m >>= 1) {
                ds += __shfl_xor(ds, m, 32);
                ys += __shfl_xor(ys, m, 32);
            }
            if (lane == 0) {
                acc[wave][s][0] += ds;
                acc[wave][s][1] += ys;
            }
        }
    }
    __syncthreads();
    // Fixed-order reduction across waves, write per-block partials
    for (int t = tid; t < NSTEPS * 2; t += 256) {
        const int s = t >> 1, c = t & 1;
        float sum = 0.0f;
#pragma unroll
        for (int w = 0; w < PH1_WAVES; ++w) sum += acc[w][s][c];
        part[blockIdx.x * (NSTEPS * 2) + t] = sum;
    }
}

// Phase 2: single block — fixed-order sum over blocks, then find first converged step.
__global__ __launch_bounds__(256) void pcn_phase2(const float* __restrict__ part,
                                                  float* __restrict__ sums,
                                                  int* __restrict__ kout) {
    __shared__ float s_sums[NSTEPS * 2];
    const int t = threadIdx.x;
    if (t < NSTEPS * 2) {
        float s = 0.0f;
        for (int b = 0; b < PH1_BLOCKS; ++b) s += part[b * (NSTEPS * 2) + t];
        s_sums[t] = s;
        sums[t] = s;
    }
    __syncthreads();
    if (t == 0) {
        int k = NSTEPS - 1;
        for (int s = 0; s < NSTEPS; ++s) {
            const float err = sqrtf(s_sums[2 * s]) / (sqrtf(s_sums[2 * s + 1]) + 1e-10f);
            if (err < TOL_C) { k = s; break; }
        }
        *kout = k;
    }
}

// Phase 3: re-run trajectory for k*+1 steps and write Y.
__global__ __launch_bounds__(256) void pcn_phase3(const float* __restrict__ X,
                                                  const float* __restrict__ W,
                                                  const float* __restrict__ Mm,
                                                  const float* __restrict__ bv,
                                                  const int* __restrict__ kptr,
                                                  float* __restrict__ out) {
    const int tid  = threadIdx.x;
    const int lane = tid & 31;
    const int wave = tid >> 5;
    const int half = lane >> 4;
    const int l15  = lane & 15;
    const int n0 = l15;
    const int n1 = l15 + 16;
    const bool v1 = (n1 < DOUT);
    const int rowBase = (blockIdx.x * 8 + wave) * 16;

    v2f b0, b1;
    float bb0, bb1, dm0, dm1, rden0, rden1;
    make_lane_consts(W, Mm, bv, lane, b0, b1, bb0, bb1, dm0, dm1, rden0, rden1);

    v8f d0, d1;
    compute_drive(X, rowBase, lane, b0, b1, bb0, bb1, d0, d1);

    const int nsteps = *kptr + 1;

    float uy0[8], yo0[8], uy1[8], yo1[8];
#pragma unroll
    for (int r = 0; r < 8; ++r) { uy0[r] = 0.f; yo0[r] = 0.f; uy1[r] = 0.f; yo1[r] = 0.f; }

    for (int s = 0; s < nsteps; ++s) {
#pragma unroll
        for (int r = 0; r < 8; ++r) {
            float du0 = d0[r] - uy0[r] - yo0[r] * dm0;
            uy0[r] += DT_C * du0;
            yo0[r] = fmaxf((uy0[r] - LBD1_C) * rden0, 0.0f);

            float du1 = d1[r] - uy1[r] - yo1[r] * dm1;
            uy1[r] += DT_C * du1;
            yo1[r] = fmaxf((uy1[r] - LBD1_C) * rden1, 0.0f);
        }
    }

#pragma unroll
    for (int r = 0; r < 8; ++r) {
        const int row = rowBase + half * 8 + r;
        out[row * DOUT + n0] = yo0[r];
        if (v1) out[row * DOUT + n1] = yo1[r];
    }
}

extern "C" void kernel_launch(void* const* d_in, const int* in_sizes, int n_in,
                              void* d_out, int out_size, void* d_ws, size_t ws_size,
                              hipStream_t stream) {
    const float* X  = (const float*)d_in[0];  // [N, 3]
    const float* W  = (const float*)d_in[1];  // [20, 3]
    const float* Mm = (const float*)d_in[2];  // [20, 20]
    const float* bv = (const float*)d_in[3];  // [20]

    float* part = (float*)d_ws;                       // PH1_BLOCKS * 200 floats
    float* sums = part + PART_FLOATS;                 // 200 floats
    int*   kptr = (int*)(sums + NSTEPS * 2);          // 1 int

    pcn_phase1<<<PH1_BLOCKS, 256, 0, stream>>>(X, W, Mm, bv, part);
    pcn_phase2<<<1, 256, 0, stream>>>(part, sums, kptr);
    pcn_phase3<<<PH3_BLOCKS, 256, 0, stream>>>(X, W, Mm, bv, kptr, (float*)d_out);
}